// StackedAffinityNet_45732811767828
// MI455X (gfx1250) — compile-verified
//
#include <hip/hip_runtime.h>
#include <math.h>

// ---------------------------------------------------------------------------
// StackedAffinityNet on MI455X (gfx1250), wave32, f32 WMMA (16x16x4).
//
//  o = _affinity(x, w, w2, k=32):
//    scores s_ij = x_i.x_j - 0.5||x_j||^2  (argmax set == argmin d2 set)
//    idx = top32_j(s_ij)                   (fused WMMA + LDS top-k, no d2 matrix)
//    A = X@Wn^T, B = X@Ws^T  (W=[Wn|Ws]) ; pooled_i = mean_k clip(A[idx]+B_i)
//    o = pooled@w2a^T + x@w2b^T           (w2 = [w2a | w2b])
// ---------------------------------------------------------------------------

typedef __attribute__((ext_vector_type(2))) float v2f;
typedef __attribute__((ext_vector_type(8))) float v8f;

#define NROWS 16384
#define NEG_INF (-3.0e38f)

// D = A(16x4,f32) x B(4x16,f32) + C, wave32.
// A frag: lane l -> row m=l&15, k pair kq=2*(l>>4)   (VGPR0=K even, VGPR1=K odd)
// B frag: lane l -> col n=l&15, same k pair          (B[k][n] = W[n][k])
// C/D   : VGPR v, lane l -> row v + 8*(l>>4), col l&15
static __device__ __forceinline__ v8f wmma_f32_4(v2f a, v2f b, v8f c) {
  return __builtin_amdgcn_wmma_f32_16x16x4_f32(false, a, false, b, (short)0, c,
                                               false, false);
}

// ---------------------------------------------------------------------------
// 0.5 * ||row||^2
__global__ void san_halfsq(const float* __restrict__ X, float* __restrict__ hs,
                           int D) {
  int i = blockIdx.x * blockDim.x + threadIdx.x;
  if (i < NROWS) {
    const float* r = X + (size_t)i * D;
    float s = 0.f;
    for (int d = 0; d < D; ++d) s += r[d] * r[d];
    hs[i] = 0.5f * s;
  }
}

// ---------------------------------------------------------------------------
// Fused score GEMM (f32 WMMA) + top-32 selection.
// Block: 128 threads (4 waves) -> 16 query rows. Wave w owns cols [16w,16w+16).
__global__ void __launch_bounds__(128) san_topk(const float* __restrict__ X,
                                                const float* __restrict__ hs,
                                                int D, int* __restrict__ idx_out) {
  __shared__ float sXq[16 * 128];      // query tile (row-major, up to D=128)
  __shared__ float sScore[16][64];     // 16 rows x 64 candidate cols
  __shared__ float sVal[16][32];       // top-k values (unsorted)
  __shared__ int sIdx[16][32];         // top-k indices
  __shared__ float sThresh[16];        // current 32nd-best score

  const int tid = threadIdx.x;
  const int wave = tid >> 5;
  const int lane = tid & 31;
  const int qbase = blockIdx.x * 16;
  const int m = lane & 15;             // row/col within 16-tile
  const int kq = (lane >> 4) * 2;      // lane's k-pair offset

  for (int t = tid; t < 16 * D; t += 128)
    sXq[t] = X[(size_t)(qbase + t / D) * D + (t % D)];
  for (int t = tid; t < 16 * 32; t += 128) {
    sVal[t / 32][t & 31] = NEG_INF;
    sIdx[t / 32][t & 31] = 0;
  }
  if (tid < 16) sThresh[tid] = NEG_INF;
  __syncthreads();

  for (int jt = 0; jt < NROWS; jt += 64) {
    const int jbase = jt + wave * 16;
    // B operand base: candidate row (jbase+m), this lane's k pair
    const float* bp = X + (size_t)(jbase + m) * D + kq;
    const float* ap = sXq + m * D + kq;

    v8f acc = {};
#pragma unroll 8
    for (int k = 0; k < D; k += 4) {
      float2 af = *(const float2*)(ap + k);
      float2 bf = *(const float2*)(bp + k);
      v2f a;
      a.x = af.x;
      a.y = af.y;
      v2f b;
      b.x = bf.x;
      b.y = bf.y;
      acc = wmma_f32_4(a, b, acc);
    }

    const float bias = hs[jbase + m];  // 0.5||x_j||^2 for col j = jbase + m
    const int rq = 8 * (lane >> 4);
#pragma unroll
    for (int v = 0; v < 8; ++v) sScore[rq + v][wave * 16 + m] = acc[v] - bias;
    __syncthreads();

    // 16 owner threads scan 64 candidates each; rare-insert threshold filter.
    if (tid < 16) {
      const int r = tid;
      float th = sThresh[r];
      for (int c = 0; c < 64; ++c) {
        float s = sScore[r][c];
        if (s > th) {
          int mp = 0;
          float mv = sVal[r][0];
          for (int p = 1; p < 32; ++p)
            if (sVal[r][p] < mv) { mv = sVal[r][p]; mp = p; }
          sVal[r][mp] = s;
          sIdx[r][mp] = jt + c;
          mv = sVal[r][0];
          for (int p = 1; p < 32; ++p) mv = fminf(mv, sVal[r][p]);
          th = mv;
        }
      }
      sThresh[r] = th;
    }
    __syncthreads();
  }

  for (int t = tid; t < 16 * 32; t += 128)
    idx_out[(size_t)(qbase + t / 32) * 32 + (t & 31)] = sIdx[t / 32][t & 31];
}

// ---------------------------------------------------------------------------
// C[16x16 tile] = U(N x K, ldu) @ W^T  (W is (cols x K), ldw), f32 WMMA.
// One wave per tile. accum!=0 -> C += result. C has row stride ldc.
__global__ void __launch_bounds__(32) san_gemm_xt(const float* __restrict__ U,
                                                  int ldu,
                                                  const float* __restrict__ W,
                                                  int ldw, float* __restrict__ C,
                                                  int ldc, int K, int accum) {
  const int lane = threadIdx.x;
  const int m = lane & 15;
  const int kq = (lane >> 4) * 2;
  const int rbase = blockIdx.x * 16;
  const int cbase = blockIdx.y * 16;

  const float* up = U + (size_t)(rbase + m) * ldu + kq;
  const float* wp = W + (size_t)(cbase + m) * ldw + kq;

  v8f acc = {};
#pragma unroll 8
  for (int k = 0; k < K; k += 4) {
    float2 af = *(const float2*)(up + k);
    float2 bf = *(const float2*)(wp + k);
    v2f a;
    a.x = af.x;
    a.y = af.y;
    v2f b;
    b.x = bf.x;
    b.y = bf.y;
    acc = wmma_f32_4(a, b, acc);
  }

  const int rq = 8 * (lane >> 4);
#pragma unroll
  for (int v = 0; v < 8; ++v) {
    float* cp = C + (size_t)(rbase + rq + v) * ldc + cbase + m;
    *cp = accum ? (*cp + acc[v]) : acc[v];
  }
}

// ---------------------------------------------------------------------------
// pooled[i][h] = mean_{n<32} clip(A[idx[i][n]][h] + B[i][h], -1, 1); H == 128.
__global__ void __launch_bounds__(128) san_pool(const float* __restrict__ A,
                                                const float* __restrict__ B,
                                                const int* __restrict__ idx,
                                                float* __restrict__ pooled) {
  const int i = blockIdx.x;
  const int h = threadIdx.x;
  const float b = B[(size_t)i * 128 + h];
  float s = 0.f;
#pragma unroll 4
  for (int n = 0; n < 32; ++n) {
    int j = idx[(size_t)i * 32 + n];
    float v = A[(size_t)j * 128 + h] + b;
    s += fminf(1.f, fmaxf(-1.f, v));
  }
  pooled[(size_t)i * 128 + h] = s * (1.f / 32.f);
}

// x1[i][0:64] = x[i][0:64]  (second half filled by the layer-0 output GEMM)
__global__ void san_copy_x(const float* __restrict__ x, float* __restrict__ x1) {
  int t = blockIdx.x * blockDim.x + threadIdx.x;  // N*64 threads
  if (t < NROWS * 64) x1[(size_t)(t >> 6) * 128 + (t & 63)] = x[t];
}

// ---------------------------------------------------------------------------
extern "C" void kernel_launch(void* const* d_in, const int* in_sizes, int n_in,
                              void* d_out, int out_size, void* d_ws,
                              size_t ws_size, hipStream_t stream) {
  const int N = NROWS;
  const float* x = (const float*)d_in[0];     // (N, 64)
  const float* w0 = (const float*)d_in[1];    // (128, 128) = [Wn(64) | Ws(64)]
  const float* w2_0 = (const float*)d_in[2];  // (64, 192)  = [pool(128) | x(64)]
  const float* w1 = (const float*)d_in[3];    // (128, 256) = [Wn(128)| Ws(128)]
  const float* w2_1 = (const float*)d_in[4];  // (64, 256)  = [pool(128)|x1(128)]
  float* out = (float*)d_out;                 // (N, 64)

  float* ws = (float*)d_ws;
  float* hs = ws;                   // N
  float* Abuf = hs + N;             // N*128
  float* Bbuf = Abuf + (size_t)N * 128;
  float* pooled = Bbuf + (size_t)N * 128;
  float* x1 = pooled + (size_t)N * 128;  // N*128  ([x | o0])
  int* idx = (int*)(x1 + (size_t)N * 128);  // N*32

  const dim3 blk128(128), blk256(256), blk32(32);
  const dim3 gTop(N / 16);
  const dim3 gGemmH(N / 16, 128 / 16);  // N x 128 outputs
  const dim3 gGemmO(N / 16, 64 / 16);   // N x 64 outputs

  // ---------------- Layer 0 (D = 64) ----------------
  san_halfsq<<<(N + 255) / 256, blk256, 0, stream>>>(x, hs, 64);
  san_topk<<<gTop, blk128, 0, stream>>>(x, hs, 64, idx);
  // A = X @ Wn^T, B = X @ Ws^T   (Wn = w0[:, :64], Ws = w0[:, 64:])
  san_gemm_xt<<<gGemmH, blk32, 0, stream>>>(x, 64, w0, 128, Abuf, 128, 64, 0);
  san_gemm_xt<<<gGemmH, blk32, 0, stream>>>(x, 64, w0 + 64, 128, Bbuf, 128, 64, 0);
  san_pool<<<N, blk128, 0, stream>>>(Abuf, Bbuf, idx, pooled);
  san_copy_x<<<(N * 64 + 255) / 256, blk256, 0, stream>>>(x, x1);
  // o0 = pooled @ w2_0[:, :128]^T + x @ w2_0[:, 128:]^T  -> x1[:, 64:128]
  san_gemm_xt<<<gGemmO, blk32, 0, stream>>>(pooled, 128, w2_0, 192, x1 + 64, 128,
                                            128, 0);
  san_gemm_xt<<<gGemmO, blk32, 0, stream>>>(x, 64, w2_0 + 128, 192, x1 + 64, 128,
                                            64, 1);

  // ---------------- Layer 1 (D = 128) ----------------
  san_halfsq<<<(N + 255) / 256, blk256, 0, stream>>>(x1, hs, 128);
  san_topk<<<gTop, blk128, 0, stream>>>(x1, hs, 128, idx);
  san_gemm_xt<<<gGemmH, blk32, 0, stream>>>(x1, 128, w1, 256, Abuf, 128, 128, 0);
  san_gemm_xt<<<gGemmH, blk32, 0, stream>>>(x1, 128, w1 + 128, 256, Bbuf, 128,
                                            128, 0);
  san_pool<<<N, blk128, 0, stream>>>(Abuf, Bbuf, idx, pooled);
  // o1 = pooled @ w2_1[:, :128]^T + x1 @ w2_1[:, 128:]^T -> d_out
  san_gemm_xt<<<gGemmO, blk32, 0, stream>>>(pooled, 128, w2_1, 256, out, 64, 128,
                                            0);
  san_gemm_xt<<<gGemmO, blk32, 0, stream>>>(x1, 128, w2_1 + 128, 256, out, 64,
                                            128, 1);
}